// HypAgg_38242388803704
// MI455X (gfx1250) — compile-verified
//
#include <hip/hip_runtime.h>
#include <hip/hip_bf16.h>

typedef __attribute__((ext_vector_type(2))) float v2f;
typedef __attribute__((ext_vector_type(8))) float v8f;

#define DIN 128
#define DOUT 16

// ---------------------------------------------------------------------------
// Kernel 1: x_tangent = logmap0(x, c=1): x / ||x|| * artanh(clip(||x||))
// One wave32 per node row (128 floats -> 4 per lane via float4).
// ---------------------------------------------------------------------------
__global__ void __launch_bounds__(256)
tangent_kernel(const float* __restrict__ x, float* __restrict__ xt, int n)
{
    int gid  = blockIdx.x * blockDim.x + threadIdx.x;
    int row  = gid >> 5;          // one wave per row
    int lane = threadIdx.x & 31;
    if (row >= n) return;

    const float4 v = *(const float4*)(x + (size_t)row * DIN + lane * 4);
    float s = v.x * v.x + v.y * v.y + v.z * v.z + v.w * v.w;
    s += __shfl_xor(s, 1, 32);
    s += __shfl_xor(s, 2, 32);
    s += __shfl_xor(s, 4, 32);
    s += __shfl_xor(s, 8, 32);
    s += __shfl_xor(s, 16, 32);

    float norm = sqrtf(s);
    float xn   = fmaxf(norm, 1e-15f);               // MIN_NORM
    float xcl  = fminf(xn, 1.0f - 1e-7f);           // artanh clip
    float at   = 0.5f * logf((1.0f + xcl) / (1.0f - xcl));
    float sc   = at / xn;

    float4 o = make_float4(v.x * sc, v.y * sc, v.z * sc, v.w * sc);
    *(float4*)(xt + (size_t)row * DIN + lane * 4) = o;
}

// ---------------------------------------------------------------------------
// Kernel 2: support = xt @ W_r   ([N x 128] @ [128 x 16]) via
// V_WMMA_F32_16X16X4_F32. 4 waves per block, one 16-row tile per wave.
// A tile and B staged through LDS with coalesced float4 loads.
// ---------------------------------------------------------------------------
__global__ void __launch_bounds__(128)
gemm_wmma_kernel(const float* __restrict__ xt, const float* __restrict__ W,
                 float* __restrict__ support, int n)
{
    __shared__ float sA[4 * 16 * DIN];   // 32 KB
    __shared__ float sB[DIN * DOUT];     //  8 KB

    const int tid  = threadIdx.x;
    const int wave = tid >> 5;
    const int lane = tid & 31;
    const int tileBase = blockIdx.x * 64 + wave * 16;

    // stage B: 2048 floats, all 128 threads
    for (int i = tid; i < DIN * DOUT; i += 128) sB[i] = W[i];

    // stage A tile for this wave: 16 rows x 128 cols = 512 float4
    float* sAw = &sA[wave * 16 * DIN];
    #pragma unroll
    for (int i = 0; i < 16; ++i) {
        int idx = i * 32 + lane;            // 0..511
        int row = idx >> 5;                 // 0..15
        int c4  = (idx & 31) << 2;          // 0..124
        int g   = tileBase + row;
        float4 val = (g < n) ? *(const float4*)(xt + (size_t)g * DIN + c4)
                             : make_float4(0.f, 0.f, 0.f, 0.f);
        *(float4*)(sAw + row * DIN + c4) = val;
    }
    __syncthreads();

    const int half = lane >> 4;   // 0 = lanes 0-15, 1 = lanes 16-31
    const int l16  = lane & 15;

    v8f c = {};
    #pragma unroll
    for (int kk = 0; kk < DIN; kk += 4) {
        int ka = kk + half * 2;   // A: lanes 0-15 hold K0/K1, lanes 16-31 K2/K3
        v2f a, b;
        a.x = sAw[l16 * DIN + ka + 0];
        a.y = sAw[l16 * DIN + ka + 1];
        // B (4x16, row striped across lanes per VGPR, like C layout):
        b.x = sB[(ka + 0) * DOUT + l16];
        b.y = sB[(ka + 1) * DOUT + l16];
        c = __builtin_amdgcn_wmma_f32_16x16x4_f32(
                false, a, false, b, (short)0, c, false, false);
    }

    // C layout: VGPR v -> row M = v + half*8, col = l16
    #pragma unroll
    for (int v = 0; v < 8; ++v) {
        int row = tileBase + v + half * 8;
        if (row < n) support[(size_t)row * DOUT + l16] = c[v];
    }
}

// ---------------------------------------------------------------------------
// Kernel 3: edge gather/scale/scatter-add. 16 lanes per edge (one per dim).
// ---------------------------------------------------------------------------
__global__ void __launch_bounds__(256)
edge_kernel(const float* __restrict__ support, const int* __restrict__ rows,
            const int* __restrict__ cols, const float* __restrict__ vals,
            float* __restrict__ agg, int e_cnt)
{
    int gid = blockIdx.x * blockDim.x + threadIdx.x;
    int e = gid >> 4;
    int d = gid & 15;
    if (e >= e_cnt) return;
    int   cc = cols[e];
    int   rr = rows[e];
    float vv = vals[e];
    float g  = support[(size_t)cc * DOUT + d] * vv;
    atomicAdd(&agg[(size_t)rr * DOUT + d], g);
}

// ---------------------------------------------------------------------------
// Kernel 4: acc += proj(expmap0(agg)).  16 lanes per node.
// ---------------------------------------------------------------------------
__global__ void __launch_bounds__(256)
node_accum_kernel(const float* __restrict__ agg, float* __restrict__ acc, int n)
{
    int gid  = blockIdx.x * blockDim.x + threadIdx.x;
    int node = gid >> 4;
    int d    = gid & 15;
    if (node >= n) return;

    float u = agg[(size_t)node * DOUT + d];
    float s = u * u;
    s += __shfl_xor(s, 1, 32);
    s += __shfl_xor(s, 2, 32);
    s += __shfl_xor(s, 4, 32);
    s += __shfl_xor(s, 8, 32);

    float un  = sqrtf(s);
    float unc = fmaxf(un, 1e-15f);          // MIN_NORM
    float t   = tanhf(unc);                 // sqrt_c = 1
    float y   = t * u / unc;                // expmap0
    float yn  = fmaxf(t * (un / unc), 1e-15f);
    const float maxnorm = 1.0f - 4e-3f;     // PROJ_EPS
    if (yn > maxnorm) y = y * (maxnorm / yn);

    acc[(size_t)node * DOUT + d] += y;
}

// ---------------------------------------------------------------------------
// Kernel 5: out = acc / 11 + bias
// ---------------------------------------------------------------------------
__global__ void __launch_bounds__(256)
finalize_kernel(float* __restrict__ out, const float* __restrict__ bias, int total)
{
    int gid = blockIdx.x * blockDim.x + threadIdx.x;
    if (gid >= total) return;
    out[gid] = out[gid] * (1.0f / 11.0f) + bias[gid & 15];
}

// ---------------------------------------------------------------------------
extern "C" void kernel_launch(void* const* d_in, const int* in_sizes, int n_in,
                              void* d_out, int out_size, void* d_ws, size_t ws_size,
                              hipStream_t stream)
{
    const float* x    = (const float*)d_in[0];   // [N,128]
    const int*   rows = (const int*)  d_in[1];   // [11,E]
    const int*   cols = (const int*)  d_in[2];   // [11,E]
    const float* vals = (const float*)d_in[3];   // [11,E]
    const float* W    = (const float*)d_in[4];   // [11,128,16]
    const float* bias = (const float*)d_in[5];   // [16]

    const int NADJ = 11;
    const int N = in_sizes[0] / DIN;
    const int E = in_sizes[1] / NADJ;

    float* xt      = (float*)d_ws;                 // N*128 f32
    float* support = xt + (size_t)N * DIN;         // N*16  f32
    float* agg     = support + (size_t)N * DOUT;   // N*16  f32

    hipMemsetAsync(d_out, 0, (size_t)out_size * sizeof(float), stream);

    tangent_kernel<<<(N + 7) / 8, 256, 0, stream>>>(x, xt, N);

    for (int r = 0; r < NADJ; ++r) {
        hipMemsetAsync(agg, 0, (size_t)N * DOUT * sizeof(float), stream);
        gemm_wmma_kernel<<<(N + 63) / 64, 128, 0, stream>>>(
            xt, W + (size_t)r * DIN * DOUT, support, N);
        edge_kernel<<<((size_t)E * 16 + 255) / 256, 256, 0, stream>>>(
            support, rows + (size_t)r * E, cols + (size_t)r * E,
            vals + (size_t)r * E, agg, E);
        node_accum_kernel<<<((size_t)N * DOUT + 255) / 256, 256, 0, stream>>>(
            agg, (float*)d_out, N);
    }

    finalize_kernel<<<((size_t)N * DOUT + 255) / 256, 256, 0, stream>>>(
        (float*)d_out, bias, N * DOUT);
}